// WindowAttention_2774548873750
// MI455X (gfx1250) — compile-verified
//
#include <hip/hip_runtime.h>
#include <hip/hip_bf16.h>

typedef __attribute__((ext_vector_type(2))) float v2f;
typedef __attribute__((ext_vector_type(8))) float v8f;
typedef __attribute__((ext_vector_type(4))) int   v4i;

#define NWIN 8192
#define NTOK 64
#define NHEAD 3
#define HDIM 32
#define SCALE 0.17677669529663687f   // 32^-0.5
#define QSTR 36                       // padded LDS row stride for q/k/v
#define PSTR 68                       // padded LDS row stride for P slab

#if defined(__AMDGCN__) && __has_builtin(__builtin_amdgcn_global_load_async_to_lds_b128)
#define USE_ASYNC_LDS 1
#else
#define USE_ASYNC_LDS 0
#endif

// ---------------------------------------------------------------------------
// Kernel 1: continuous relative position bias MLP:
// bias[h][m][n] = relu(rel_pos[m,n,:] @ W1 + b1) @ W2 + b2   (h=0..2, m,n=0..63)
// ---------------------------------------------------------------------------
__global__ void crpb_bias_kernel(const float* __restrict__ rel_pos,
                                 const float* __restrict__ W1,
                                 const float* __restrict__ b1,
                                 const float* __restrict__ W2,
                                 const float* __restrict__ b2,
                                 float* __restrict__ bias) {
    int idx = blockIdx.x * blockDim.x + threadIdx.x;   // m*64 + n, 0..4095
    if (idx >= NTOK * NTOK) return;
    float r0 = rel_pos[idx * 2 + 0];
    float r1 = rel_pos[idx * 2 + 1];
    float a0 = 0.f, a1 = 0.f, a2 = 0.f;
#pragma unroll 4
    for (int j = 0; j < 256; ++j) {
        float hv = fmaxf(fmaf(r0, W1[j], fmaf(r1, W1[256 + j], b1[j])), 0.f);
        a0 = fmaf(hv, W2[j * 3 + 0], a0);
        a1 = fmaf(hv, W2[j * 3 + 1], a1);
        a2 = fmaf(hv, W2[j * 3 + 2], a2);
    }
    bias[0 * 4096 + idx] = a0 + b2[0];
    bias[1 * 4096 + idx] = a1 + b2[1];
    bias[2 * 4096 + idx] = a2 + b2[2];
}

// ---------------------------------------------------------------------------
// Kernel 2: one block (4 waves, 128 threads) per (head, window) pair.
// Each wave handles a 16-row slab of the 64x64 attention matrix.
// ---------------------------------------------------------------------------
__global__ __launch_bounds__(128)
void window_attn_kernel(const float* __restrict__ qkv,
                        const float* __restrict__ mask,
                        const float* __restrict__ bias,
                        float* __restrict__ out) {
    __shared__ float q_s[NTOK * QSTR];
    __shared__ float k_s[NTOK * QSTR];
    __shared__ float v_s[NTOK * QSTR];
    __shared__ float p_s[4 * 16 * PSTR];

    const int tid   = threadIdx.x;
    const int lane  = tid & 31;
    const int wave  = tid >> 5;
    const int l16   = lane & 15;
    const int khalf = (lane >> 4) << 1;      // 0 for lanes 0-15, 2 for lanes 16-31
    const int mloc  = (lane < 16) ? 0 : 8;   // C/D tile row offset per half-wave
    const int m0    = wave * 16;             // slab row base
    const int h     = blockIdx.x;            // head 0..2
    const int win   = blockIdx.y;            // window 0..8191
    const int wmask = win & 1023;            // window index within image (nW=1024)

    const float* qkv_w  = qkv + (size_t)win * NTOK * (3 * NHEAD * HDIM);
    const float* mask_w = mask + (size_t)wmask * NTOK * NTOK;
    const float* bias_h = bias + h * NTOK * NTOK;
    float*       out_w  = out + (size_t)win * NTOK * (NHEAD * HDIM) + h * HDIM;
    float*       pw     = p_s + wave * 16 * PSTR;

    // ---- load Q/K/V [64 x 32] head slices into padded LDS.
    // Each 8-lane group covers one token's 32 floats = one 128B-aligned line.
#if USE_ASYNC_LDS
    for (int i = tid; i < 3 * NTOK * 8; i += 128) {
        int arr = i >> 9;            // 0..2  (q,k,v)
        int n   = (i >> 3) & 63;     // token
        int dg  = i & 7;             // float4 within head dim
        const float* g = qkv_w + n * 288 + arr * 96 + h * 32 + dg * 4;
        float* l = (arr == 0 ? q_s : arr == 1 ? k_s : v_s) + n * QSTR + dg * 4;
        __builtin_amdgcn_global_load_async_to_lds_b128(
            (__attribute__((address_space(1))) v4i*)g,
            (__attribute__((address_space(3))) v4i*)l,
            0, 0);
    }
    __builtin_amdgcn_s_wait_asynccnt(0);
#else
    {
        const float4* src = (const float4*)qkv_w;
        for (int i = tid; i < 3 * NTOK * 8; i += 128) {
            int arr = i >> 9;
            int n   = (i >> 3) & 63;
            int dg  = i & 7;
            float4 val = src[n * 72 + arr * 24 + h * 8 + dg];
            float* dst = (arr == 0 ? q_s : arr == 1 ? k_s : v_s) + n * QSTR + dg * 4;
            dst[0] = val.x; dst[1] = val.y; dst[2] = val.z; dst[3] = val.w;
        }
    }
#endif
    __syncthreads();

    // ---- S = (Q*scale) @ K^T : 4 tiles of 16x16, k-depth 32 in steps of 4
    v8f acc[4];
#pragma unroll
    for (int t = 0; t < 4; ++t) acc[t] = (v8f){0.f,0.f,0.f,0.f,0.f,0.f,0.f,0.f};

#pragma unroll
    for (int kk = 0; kk < 8; ++kk) {
        int klo = kk * 4 + khalf;
        v2f a;
        a.x = q_s[(m0 + l16) * QSTR + klo]     * SCALE;
        a.y = q_s[(m0 + l16) * QSTR + klo + 1] * SCALE;
#pragma unroll
        for (int t = 0; t < 4; ++t) {
            int col = t * 16 + l16;
            v2f b;
            b.x = k_s[col * QSTR + klo];
            b.y = k_s[col * QSTR + klo + 1];
            acc[t] = __builtin_amdgcn_wmma_f32_16x16x4_f32(
                false, a, false, b, (short)0, acc[t], false, false);
        }
    }

    // ---- add bias + mask, softmax over 64 cols (16 lanes x 4 tiles)
#pragma unroll
    for (int vr = 0; vr < 8; ++vr) {
        int m = m0 + vr + mloc;                  // global query row
        float vals[4];
        float rmax = -3.402823466e38f;
#pragma unroll
        for (int t = 0; t < 4; ++t) {
            int col = t * 16 + l16;
            float s = acc[t][vr] + bias_h[m * 64 + col] + mask_w[m * 64 + col];
            vals[t] = s;
            rmax = fmaxf(rmax, s);
        }
#pragma unroll
        for (int off = 1; off < 16; off <<= 1)
            rmax = fmaxf(rmax, __shfl_xor(rmax, off, 32));
        float rsum = 0.f;
#pragma unroll
        for (int t = 0; t < 4; ++t) {
            float p = __expf(vals[t] - rmax);
            vals[t] = p;
            rsum += p;
        }
#pragma unroll
        for (int off = 1; off < 16; off <<= 1)
            rsum += __shfl_xor(rsum, off, 32);
        float inv = 1.f / rsum;
#pragma unroll
        for (int t = 0; t < 4; ++t) acc[t][vr] = vals[t] * inv;
    }

    // ---- spill P slab [16 x 64] to this wave's private LDS strip
    // (same-wave LDS ops are hardware in-order; compiler barrier only)
#pragma unroll
    for (int vr = 0; vr < 8; ++vr) {
        int mr = vr + mloc;
#pragma unroll
        for (int t = 0; t < 4; ++t)
            pw[mr * PSTR + t * 16 + l16] = acc[t][vr];
    }
    __asm__ volatile("" ::: "memory");

    // ---- O = P @ V : 2 tiles of 16x16, k-depth 64 in steps of 4
    v8f o0 = (v8f){0.f,0.f,0.f,0.f,0.f,0.f,0.f,0.f};
    v8f o1 = o0;
#pragma unroll
    for (int kk = 0; kk < 16; ++kk) {
        int klo = kk * 4 + khalf;
        v2f a;
        a.x = pw[l16 * PSTR + klo];
        a.y = pw[l16 * PSTR + klo + 1];
        v2f b0, b1v;
        b0.x  = v_s[klo * QSTR + l16];
        b0.y  = v_s[(klo + 1) * QSTR + l16];
        b1v.x = v_s[klo * QSTR + 16 + l16];
        b1v.y = v_s[(klo + 1) * QSTR + 16 + l16];
        o0 = __builtin_amdgcn_wmma_f32_16x16x4_f32(
            false, a, false, b0, (short)0, o0, false, false);
        o1 = __builtin_amdgcn_wmma_f32_16x16x4_f32(
            false, a, false, b1v, (short)0, o1, false, false);
    }

    // ---- store O slab [16 x 32] to x[win, n, h*32 + d]
#pragma unroll
    for (int vr = 0; vr < 8; ++vr) {
        int n = m0 + vr + mloc;
        out_w[(size_t)n * (NHEAD * HDIM) + l16]      = o0[vr];
        out_w[(size_t)n * (NHEAD * HDIM) + 16 + l16] = o1[vr];
    }
}

// ---------------------------------------------------------------------------
extern "C" void kernel_launch(void* const* d_in, const int* in_sizes, int n_in,
                              void* d_out, int out_size, void* d_ws, size_t ws_size,
                              hipStream_t stream) {
    const float* qkv     = (const float*)d_in[0];
    const float* mask    = (const float*)d_in[1];
    const float* W1      = (const float*)d_in[2];
    const float* b1      = (const float*)d_in[3];
    const float* W2      = (const float*)d_in[4];
    const float* b2      = (const float*)d_in[5];
    const float* rel_pos = (const float*)d_in[6];
    float* out  = (float*)d_out;
    float* bias = (float*)d_ws;   // 3*64*64 floats = 48 KB

    crpb_bias_kernel<<<16, 256, 0, stream>>>(rel_pos, W1, b1, W2, b2, bias);
    window_attn_kernel<<<dim3(NHEAD, NWIN), 128, 0, stream>>>(qkv, mask, bias, out);
}